// GATNet_69045894250548
// MI455X (gfx1250) — compile-verified
//
#include <hip/hip_runtime.h>
#include <math.h>

typedef __attribute__((ext_vector_type(2))) float v2f;
typedef __attribute__((ext_vector_type(8))) float v8f;

// ---------- helpers ----------
__device__ __forceinline__ unsigned fkey(float v) {
  // monotone float -> uint mapping (order preserving) for atomicMax
  unsigned b = __float_as_uint(v);
  return (b & 0x80000000u) ? ~b : (b | 0x80000000u);
}
__device__ __forceinline__ float funkey(unsigned k) {
  unsigned b = (k & 0x80000000u) ? (k ^ 0x80000000u) : ~k;
  return __uint_as_float(b);
}

// ---------- generic fills ----------
__global__ void fill_f32(float* p, float v, int n) {
  int i = blockIdx.x * blockDim.x + threadIdx.x;
  if (i < n) p[i] = v;
}
__global__ void fill_u32(unsigned* p, unsigned v, int n) {
  int i = blockIdx.x * blockDim.x + threadIdx.x;
  if (i < n) p[i] = v;
}

// ---------- f32 WMMA GEMM: C = act(A[MxK] @ B[KxNc] + bias) ----------
// One wave per 16x16 output tile; V_WMMA_F32_16X16X4_F32, K fully unrolled.
// K and Nc are compile-time: all A/B accesses become immediate-offset loads
// off a single base register (no per-step 64-bit address arithmetic).
template <int K, int Nc>
__global__ void gemm_wmma_f32(const float* __restrict__ A, const float* __restrict__ B,
                              const float* __restrict__ bias, float* __restrict__ Cout,
                              int M, int act) {
  constexpr int  tilesN = (Nc + 15) >> 4;
  constexpr bool FULL   = (Nc % 16) == 0;   // no column masking needed
  int wave = blockIdx.x * (blockDim.x >> 5) + (threadIdx.x >> 5);
  int tilesM = (M + 15) >> 4;               // M is a multiple of 16 in this net
  if (wave >= tilesM * tilesN) return;      // whole-wave exit: EXEC all-1 for WMMA
  int tm   = wave / tilesN;
  int tn   = wave - tm * tilesN;
  int lane = threadIdx.x & 31;
  int half = lane >> 4;                     // A/B layout: lanes 16-31 hold K+2..K+3
  int l16  = lane & 15;
  int row  = tm * 16 + l16;                 // A row for this lane
  int col  = tn * 16 + l16;                 // B/C column for this lane
  bool cok  = FULL || (col < Nc);
  int  colc = FULL ? col : (cok ? col : 0); // clamped: loads always in-bounds

  const float* ap = A + (size_t)row * K + 2 * half;     // + k0 immediates
  const float* bp = B + (size_t)(2 * half) * Nc + colc; // + k0*Nc immediates (compile-time)

  v8f c = {0.f, 0.f, 0.f, 0.f, 0.f, 0.f, 0.f, 0.f};
#pragma unroll
  for (int k0 = 0; k0 < K; k0 += 4) {
    float2 av = *(const float2*)(ap + k0);              // global_load_b64, imm offset
    float  bx = bp[(size_t)k0 * Nc];                    // global_load_b32, imm offset
    float  by = bp[(size_t)k0 * Nc + Nc];
    v2f a, b;
    a.x = av.x;
    a.y = av.y;
    b.x = FULL ? bx : (cok ? bx : 0.f);                 // cndmask only when Nc%16 != 0
    b.y = FULL ? by : (cok ? by : 0.f);
    c = __builtin_amdgcn_wmma_f32_16x16x4_f32(false, a, false, b, (short)0, c,
                                              false, false);
  }

  if (cok) {
    float bv = bias ? bias[col] : 0.f;
    float* crow = Cout + (size_t)(tm * 16 + 8 * half) * Nc + col;
#pragma unroll
    for (int v = 0; v < 8; ++v) {           // C/D layout: VGPR v -> M = v + 8*half
      float val = c[v] + bv;
      if (act) val = fmaxf(val, 0.f);       // relu
      crow[(size_t)v * Nc] = val;
    }
  }
}

// Runtime-Nc fallback (only used if the class count C is not 8).
__global__ void gemm_wmma_f32_dyn(const float* __restrict__ A, const float* __restrict__ B,
                                  const float* __restrict__ bias, float* __restrict__ Cout,
                                  int M, int K, int Nc, int act) {
  int wave   = blockIdx.x * (blockDim.x >> 5) + (threadIdx.x >> 5);
  int tilesN = (Nc + 15) >> 4;
  int tilesM = (M + 15) >> 4;
  if (wave >= tilesM * tilesN) return;
  int tm = wave / tilesN, tn = wave - tm * tilesN;
  int lane = threadIdx.x & 31, half = lane >> 4, l16 = lane & 15;
  int row = tm * 16 + l16, col = tn * 16 + l16;
  bool cok = (col < Nc);
  int colc = cok ? col : 0;
  const float* ap = A + (size_t)row * K + 2 * half;
  const float* bp = B + (size_t)(2 * half) * Nc + colc;
  v8f c = {0.f, 0.f, 0.f, 0.f, 0.f, 0.f, 0.f, 0.f};
  for (int k0 = 0; k0 < K; k0 += 4) {
    float2 av = *(const float2*)ap;
    float bx = bp[0], by = bp[Nc];
    v2f a, b;
    a.x = av.x;            a.y = av.y;
    b.x = cok ? bx : 0.f;  b.y = cok ? by : 0.f;
    c = __builtin_amdgcn_wmma_f32_16x16x4_f32(false, a, false, b, (short)0, c,
                                              false, false);
    ap += 4;
    bp += (size_t)4 * Nc;
  }
  if (cok) {
    float bv = bias ? bias[col] : 0.f;
    float* crow = Cout + (size_t)(tm * 16 + 8 * half) * Nc + col;
    for (int v = 0; v < 8; ++v) {
      float val = c[v] + bv;
      if (act) val = fmaxf(val, 0.f);
      crow[(size_t)v * Nc] = val;
    }
  }
}

// ---------- alpha_s / alpha_d: per (node, head) dot over 64 features ----------
__global__ void alpha_kernel(const float* __restrict__ h, const float* __restrict__ a_src,
                             const float* __restrict__ a_dst, float* __restrict__ as_out,
                             float* __restrict__ ad_out, int n, int heads) {
  int wave = blockIdx.x * (blockDim.x >> 5) + (threadIdx.x >> 5);
  if (wave >= n * heads) return;
  int node = wave / heads, hd = wave - node * heads;
  int lane = threadIdx.x & 31;
  float2 hv = *(const float2*)(h + (size_t)node * heads * 64 + hd * 64 + lane * 2);
  float2 as = *(const float2*)(a_src + hd * 64 + lane * 2);
  float2 ad = *(const float2*)(a_dst + hd * 64 + lane * 2);
  float ss = hv.x * as.x + hv.y * as.y;
  float sd = hv.x * ad.x + hv.y * ad.y;
  for (int off = 16; off > 0; off >>= 1) {
    ss += __shfl_xor(ss, off, 32);
    sd += __shfl_xor(sd, off, 32);
  }
  if (lane == 0) { as_out[wave] = ss; ad_out[wave] = sd; }
}

// ---------- edge pass 1: e = leaky_relu(as[src]+ad[dst]); segment max over dst ----------
__global__ void edge_max_kernel(const int* __restrict__ ei, int Eorig, int Etot,
                                const float* __restrict__ as_v, const float* __restrict__ ad_v,
                                float* __restrict__ ebuf, unsigned* __restrict__ mkey,
                                int heads) {
  int idx = blockIdx.x * blockDim.x + threadIdx.x;   // over Etot*heads
  if (idx >= Etot * heads) return;
  int e = idx / heads, hd = idx - e * heads;
  int src, dst;
  if (e < Eorig) { src = ei[e]; dst = ei[Eorig + e]; }
  else           { src = dst = e - Eorig; }          // self loops
  float v = as_v[src * heads + hd] + ad_v[dst * heads + hd];
  v = (v > 0.f) ? v : 0.2f * v;                      // leaky_relu(0.2)
  ebuf[idx] = v;
  atomicMax(&mkey[dst * heads + hd], fkey(v));
}

// ---------- edge pass 2: ex = exp(e-m); z[dst]+=ex; acc[dst] += ex*h[src] ----------
// One wave per edge. HEADS=2: each lane gathers float4 (b128) of one head
// (lane<16 -> head0, lane>=16 -> head1), 4 atomics/lane. HEADS=1: float2/lane.
template <int HEADS>
__global__ void edge_acc_kernel(const int* __restrict__ ei, int Eorig, int Etot,
                                const float* __restrict__ ebuf, const unsigned* __restrict__ mkey,
                                const float* __restrict__ h, float* __restrict__ acc,
                                float* __restrict__ z) {
  int wave = blockIdx.x * (blockDim.x >> 5) + (threadIdx.x >> 5);
  if (wave >= Etot) return;
  int lane = threadIdx.x & 31;
  int src, dst;
  if (wave < Eorig) { src = ei[wave]; dst = ei[Eorig + wave]; }
  else              { src = dst = wave - Eorig; }

  if (HEADS == 2) {
    int hd = lane >> 4;                              // lane's head
    float m  = funkey(mkey[dst * 2 + hd]);
    float ex = expf(ebuf[(size_t)wave * 2 + hd] - m);
    if (lane == 0 || lane == 16)
      atomicAdd(&z[dst * 2 + hd], ex);
    // feature offset: 4 floats per lane, row = 128 floats [head0|head1]
    float4 hv = *(const float4*)(h + (size_t)src * 128 + lane * 4);
    float* ap = acc + (size_t)dst * 128 + lane * 4;
    atomicAdd(ap + 0, ex * hv.x);
    atomicAdd(ap + 1, ex * hv.y);
    atomicAdd(ap + 2, ex * hv.z);
    atomicAdd(ap + 3, ex * hv.w);
  } else {
    float m  = funkey(mkey[dst]);
    float ex = expf(ebuf[wave] - m);
    if (lane == 0) atomicAdd(&z[dst], ex);
    float2 hv = *(const float2*)(h + (size_t)src * 64 + lane * 2);
    float* ap = acc + (size_t)dst * 64 + lane * 2;
    atomicAdd(ap,     ex * hv.x);
    atomicAdd(ap + 1, ex * hv.y);
  }
}

// ---------- finalize: out = elu(acc / (z + 1e-16) + bias) (in place) ----------
__global__ void finalize_gat(float* __restrict__ acc, const float* __restrict__ z,
                             const float* __restrict__ bias, int n, int heads) {
  int total = n * heads * 64;
  int i = blockIdx.x * blockDim.x + threadIdx.x;
  if (i >= total) return;
  int c    = i % (heads * 64);
  int node = i / (heads * 64);
  int hd   = c >> 6;
  float v = acc[i] / (z[node * heads + hd] + 1e-16f) + bias[c];
  acc[i] = (v > 0.f) ? v : expm1f(v);                 // elu
}

// ---------- host-side orchestration ----------
extern "C" void kernel_launch(void* const* d_in, const int* in_sizes, int n_in,
                              void* d_out, int out_size, void* d_ws, size_t ws_size,
                              hipStream_t stream) {
  const float* x      = (const float*)d_in[0];
  const int*   ei     = (const int*)d_in[1];
  const float* W1     = (const float*)d_in[2];
  const float* a_src1 = (const float*)d_in[3];
  const float* a_dst1 = (const float*)d_in[4];
  const float* b1     = (const float*)d_in[5];
  const float* W2     = (const float*)d_in[6];
  const float* a_src2 = (const float*)d_in[7];
  const float* a_dst2 = (const float*)d_in[8];
  const float* b2     = (const float*)d_in[9];
  const float* m1w    = (const float*)d_in[10];
  const float* m1b    = (const float*)d_in[11];
  const float* m2w    = (const float*)d_in[12];
  const float* m2b    = (const float*)d_in[13];

  const int N    = in_sizes[0] / 64;    // D = 64
  const int E    = in_sizes[1] / 2;
  const int C    = in_sizes[12] / 64;   // mlp2_w is (64, C)
  const int Etot = E + N;

  // workspace layout (floats), ~146 MB total
  float*    bufA  = (float*)d_ws;                      // N*128 (h1pre / h2pre / h3)
  float*    bufB  = bufA + (size_t)N * 128;            // N*128 (acc1 -> h1out)
  float*    bufC  = bufB + (size_t)N * 128;            // N*64  (acc2 -> h2out)
  float*    as_v  = bufC + (size_t)N * 64;             // N*2
  float*    ad_v  = as_v + (size_t)N * 2;              // N*2
  unsigned* mkey  = (unsigned*)(ad_v + (size_t)N * 2); // N*2
  float*    zbuf  = (float*)mkey + (size_t)N * 2;      // N*2
  float*    ebuf  = zbuf + (size_t)N * 2;              // Etot*2

  const int TPB = 256;
  auto blocks1d = [&](long n) { return (int)((n + TPB - 1) / TPB); };
  auto gemm_blocks = [&](int M, int Nc) {
    int waves = ((M + 15) / 16) * ((Nc + 15) / 16);
    return (waves + 7) / 8;
  };
  const unsigned KEY_NEG_INF = 0x007FFFFFu;            // fkey(-inf)

  // ===== conv1: GAT(64 -> 64, heads=2, concat) =====
  gemm_wmma_f32<64, 128><<<gemm_blocks(N, 128), TPB, 0, stream>>>(x, W1, nullptr,
                                                                  bufA, N, 0); // h1pre
  alpha_kernel<<<(N * 2 + 7) / 8, TPB, 0, stream>>>(bufA, a_src1, a_dst1,
                                                    as_v, ad_v, N, 2);
  fill_u32<<<blocks1d(N * 2), TPB, 0, stream>>>(mkey, KEY_NEG_INF, N * 2);
  fill_f32<<<blocks1d(N * 2), TPB, 0, stream>>>(zbuf, 0.f, N * 2);
  fill_f32<<<blocks1d((long)N * 128), TPB, 0, stream>>>(bufB, 0.f, N * 128);
  edge_max_kernel<<<blocks1d((long)Etot * 2), TPB, 0, stream>>>(ei, E, Etot, as_v, ad_v,
                                                                ebuf, mkey, 2);
  edge_acc_kernel<2><<<(Etot + 7) / 8, TPB, 0, stream>>>(ei, E, Etot, ebuf, mkey,
                                                         bufA, bufB, zbuf);
  finalize_gat<<<blocks1d((long)N * 128), TPB, 0, stream>>>(bufB, zbuf, b1, N, 2);

  // ===== conv2: GAT(128 -> 64, heads=1) =====
  gemm_wmma_f32<128, 64><<<gemm_blocks(N, 64), TPB, 0, stream>>>(bufB, W2, nullptr,
                                                                 bufA, N, 0);  // h2pre
  alpha_kernel<<<(N + 7) / 8, TPB, 0, stream>>>(bufA, a_src2, a_dst2,
                                                as_v, ad_v, N, 1);
  fill_u32<<<blocks1d(N), TPB, 0, stream>>>(mkey, KEY_NEG_INF, N);
  fill_f32<<<blocks1d(N), TPB, 0, stream>>>(zbuf, 0.f, N);
  fill_f32<<<blocks1d((long)N * 64), TPB, 0, stream>>>(bufC, 0.f, N * 64);
  edge_max_kernel<<<blocks1d((long)Etot), TPB, 0, stream>>>(ei, E, Etot, as_v, ad_v,
                                                            ebuf, mkey, 1);
  edge_acc_kernel<1><<<(Etot + 7) / 8, TPB, 0, stream>>>(ei, E, Etot, ebuf, mkey,
                                                         bufA, bufC, zbuf);
  finalize_gat<<<blocks1d((long)N * 64), TPB, 0, stream>>>(bufC, zbuf, b2, N, 1);

  // ===== MLP head =====
  gemm_wmma_f32<64, 64><<<gemm_blocks(N, 64), TPB, 0, stream>>>(bufC, m1w, m1b,
                                                                bufA, N, 1);   // relu
  if (C == 8) {
    gemm_wmma_f32<64, 8><<<gemm_blocks(N, 8), TPB, 0, stream>>>(bufA, m2w, m2b,
                                                                (float*)d_out, N, 1);
  } else {
    gemm_wmma_f32_dyn<<<gemm_blocks(N, C), TPB, 0, stream>>>(bufA, m2w, m2b,
                                                             (float*)d_out, N, 64, C, 1);
  }
}